// SimpleGCN_55989193670847
// MI455X (gfx1250) — compile-verified
//
#include <hip/hip_runtime.h>
#include <hip/hip_bf16.h>

// ---------------------------------------------------------------------------
// 2-layer GCN (PyG GCNConv semantics) for MI455X / gfx1250, wave32.
// GEMMs use V_WMMA_F32_16X16X4_F32 (exact fp32 matrix pipe).
// Aggregation uses f32 global atomics (agg buffers fit in 192MB L2).
// ---------------------------------------------------------------------------

typedef float v2f __attribute__((ext_vector_type(2)));
typedef float v8f __attribute__((ext_vector_type(8)));

static constexpr int IN_CH  = 128;
static constexpr int HID_CH = 128;
static constexpr int OUT_CH = 64;

// ---------------- degree / normalization ----------------------------------

__global__ void k_init_deg(float* __restrict__ deg, int n) {
    int i = blockIdx.x * blockDim.x + threadIdx.x;
    if (i < n) deg[i] = 1.0f;                      // self loop
}

__global__ void k_edge_deg(const int* __restrict__ dst, int E,
                           float* __restrict__ deg) {
    int e = blockIdx.x * blockDim.x + threadIdx.x;
    if (e < E) atomicAdd(&deg[dst[e]], 1.0f);
}

__global__ void k_dinv(float* __restrict__ deg, int n) {
    int i = blockIdx.x * blockDim.x + threadIdx.x;
    if (i < n) deg[i] = rsqrtf(deg[i]);            // deg >= 1 always
}

// ---------------- fp32 WMMA GEMM: C[M,NDIM] = A[M,K] @ B[K,NDIM] -----------
// One wave computes one 16x16 output tile, K-loop of V_WMMA_F32_16X16X4_F32.
// A frag (16x4 f32): lanes 0-15 -> row=lane, K={0,1}; lanes 16-31 -> K={2,3}.
// B frag (4x16 f32): VGPR0 = rows K0/K2 (lane halves), VGPR1 = rows K1/K3.
// C/D (16x16 f32):   VGPR j = rows j (lanes 0-15) / j+8 (lanes 16-31).

template <int K, int NDIM>
__global__ void k_gemm_wmma_f32(const float* __restrict__ A,
                                const float* __restrict__ B,
                                float* __restrict__ C, int M) {
    const int lane  = threadIdx.x & 31;
    const int wave  = threadIdx.x >> 5;
    const int ntile = NDIM / 16;

    long tile  = (long)blockIdx.x * (blockDim.x >> 5) + wave;
    long total = (long)(M / 16) * ntile;
    if (tile >= total) return;                     // uniform per wave

    const int tm = (int)(tile / ntile);
    const int tn = (int)(tile % ntile);

    const int half  = lane >> 4;                   // 0 or 1
    const int l15   = lane & 15;
    const int row   = tm * 16 + l15;               // A row for this lane
    const int col   = tn * 16 + l15;               // B/C column for this lane
    const int khalf = half * 2;                    // K sub-offset {0,2}

    const float* ap = A + (long)row * K + khalf;   // a.x/a.y adjacent -> b64 load
    const float* bp = B + (long)khalf * NDIM + col;

    v8f acc = {};
#pragma unroll
    for (int k0 = 0; k0 < K; k0 += 4) {
        v2f a, b;
        a.x = ap[k0];
        a.y = ap[k0 + 1];
        b.x = bp[(long)k0 * NDIM];
        b.y = bp[(long)(k0 + 1) * NDIM];
        acc = __builtin_amdgcn_wmma_f32_16x16x4_f32(
            /*neg_a=*/false, a, /*neg_b=*/false, b,
            /*c_mod=*/(short)0, acc, /*reuse_a=*/false, /*reuse_b=*/false);
    }

    const int rbase = tm * 16 + half * 8;
#pragma unroll
    for (int j = 0; j < 8; ++j)
        C[(long)(rbase + j) * NDIM + col] = acc[j];
}

// ---------------- self-loop init: agg = dinv[i]^2 * h[i] -------------------

template <int C>
__global__ void k_self_init(const float* __restrict__ h,
                            const float* __restrict__ dinv,
                            float* __restrict__ agg, int n) {
    const int PER = C / 4;
    long idx = (long)blockIdx.x * blockDim.x + threadIdx.x;
    if (idx >= (long)n * PER) return;
    int node = (int)(idx / PER);
    int j    = (int)(idx % PER);
    float s  = dinv[node];
    s *= s;
    float4 v = ((const float4*)(h + (long)node * C))[j];
    v.x *= s; v.y *= s; v.z *= s; v.w *= s;
    ((float4*)(agg + (long)node * C))[j] = v;
}

// ---------------- edge scatter: agg[d] += dinv[s]*dinv[d]*h[s] -------------
// C/4 threads per edge; float4 gather + 4 scalar f32 atomics (L2-resident).

template <int C>
__global__ void k_edge_scatter(const int* __restrict__ src,
                               const int* __restrict__ dst,
                               const float* __restrict__ dinv,
                               const float* __restrict__ h,
                               float* __restrict__ agg, int E) {
    const int PER = C / 4;
    long idx = (long)blockIdx.x * blockDim.x + threadIdx.x;
    long e   = idx / PER;
    int  j   = (int)(idx % PER);
    if (e >= E) return;
    int s = src[e];
    int d = dst[e];
    float nm = dinv[s] * dinv[d];
    float4 v = ((const float4*)(h + (long)s * C))[j];
    float* o = agg + (long)d * C + j * 4;
    atomicAdd(o + 0, v.x * nm);
    atomicAdd(o + 1, v.y * nm);
    atomicAdd(o + 2, v.z * nm);
    atomicAdd(o + 3, v.w * nm);
}

// ---------------- epilogues ------------------------------------------------

__global__ void k_bias_relu(float* __restrict__ agg, const float* __restrict__ b,
                            long total, int C) {
    long idx = (long)blockIdx.x * blockDim.x + threadIdx.x;
    if (idx >= total) return;
    int c = (int)(idx % C);
    float v = agg[idx] + b[c];
    agg[idx] = v > 0.0f ? v : 0.0f;
}

// One wave per node: 64 logits = 2 per lane; shuffle-reduce max & sum(exp).
__global__ void k_bias_logsoftmax64(const float* __restrict__ agg,
                                    const float* __restrict__ b,
                                    float* __restrict__ out, int n) {
    const int lane = threadIdx.x & 31;
    long node = (long)blockIdx.x * (blockDim.x >> 5) + (threadIdx.x >> 5);
    if (node >= n) return;
    const float* p = agg + node * 64;
    float v0 = p[lane]      + b[lane];
    float v1 = p[lane + 32] + b[lane + 32];
    float m = fmaxf(v0, v1);
#pragma unroll
    for (int off = 16; off > 0; off >>= 1)
        m = fmaxf(m, __shfl_xor(m, off, 32));
    float s = expf(v0 - m) + expf(v1 - m);
#pragma unroll
    for (int off = 16; off > 0; off >>= 1)
        s += __shfl_xor(s, off, 32);
    float lg = logf(s);
    out[node * 64 + lane]      = v0 - m - lg;
    out[node * 64 + lane + 32] = v1 - m - lg;
}

// ---------------- launch ---------------------------------------------------

extern "C" void kernel_launch(void* const* d_in, const int* in_sizes, int n_in,
                              void* d_out, int out_size, void* d_ws, size_t ws_size,
                              hipStream_t stream) {
    const float* x  = (const float*)d_in[0];
    const int*   ei = (const int*)d_in[1];
    const float* W1 = (const float*)d_in[2];
    const float* b1 = (const float*)d_in[3];
    const float* W2 = (const float*)d_in[4];
    const float* b2 = (const float*)d_in[5];
    float* out = (float*)d_out;

    const int n = in_sizes[0] / IN_CH;     // 100000
    const int E = in_sizes[1] / 2;         // 1600000
    const int* srcp = ei;
    const int* dstp = ei + E;

    // workspace carve-out (256B aligned)
    char* ws = (char*)d_ws;
    auto carve = [&](size_t bytes) {
        float* p = (float*)ws;
        ws += (bytes + 255) & ~(size_t)255;
        return p;
    };
    float* dinv = carve((size_t)n * sizeof(float));
    float* h1   = carve((size_t)n * HID_CH * sizeof(float));
    float* agg1 = carve((size_t)n * HID_CH * sizeof(float));
    float* h2   = carve((size_t)n * OUT_CH * sizeof(float));
    float* agg2 = carve((size_t)n * OUT_CH * sizeof(float));

    const int BT = 256;
    auto cdiv = [](long a, long b) { return (int)((a + b - 1) / b); };

    // normalization
    k_init_deg<<<cdiv(n, BT), BT, 0, stream>>>(dinv, n);
    k_edge_deg<<<cdiv(E, BT), BT, 0, stream>>>(dstp, E, dinv);
    k_dinv<<<cdiv(n, BT), BT, 0, stream>>>(dinv, n);

    // ---- layer 1: h1 = x @ W1 ; agg1 = D^-1/2 A D^-1/2 h1 ; relu(+b1) ----
    {
        long tiles = (long)(n / 16) * (HID_CH / 16);
        k_gemm_wmma_f32<IN_CH, HID_CH>
            <<<cdiv(tiles, BT / 32), BT, 0, stream>>>(x, W1, h1, n);
    }
    k_self_init<HID_CH><<<cdiv((long)n * (HID_CH / 4), BT), BT, 0, stream>>>(h1, dinv, agg1, n);
    k_edge_scatter<HID_CH><<<cdiv((long)E * (HID_CH / 4), BT), BT, 0, stream>>>(
        srcp, dstp, dinv, h1, agg1, E);
    k_bias_relu<<<cdiv((long)n * HID_CH, BT), BT, 0, stream>>>(agg1, b1, (long)n * HID_CH, HID_CH);

    // ---- layer 2: h2 = relu_out @ W2 ; aggregate ; +b2 ; log_softmax -----
    {
        long tiles = (long)(n / 16) * (OUT_CH / 16);
        k_gemm_wmma_f32<HID_CH, OUT_CH>
            <<<cdiv(tiles, BT / 32), BT, 0, stream>>>(agg1, W2, h2, n);
    }
    k_self_init<OUT_CH><<<cdiv((long)n * (OUT_CH / 4), BT), BT, 0, stream>>>(h2, dinv, agg2, n);
    k_edge_scatter<OUT_CH><<<cdiv((long)E * (OUT_CH / 4), BT), BT, 0, stream>>>(
        srcp, dstp, dinv, h2, agg2, E);
    k_bias_logsoftmax64<<<cdiv(n, BT / 32), BT, 0, stream>>>(agg2, b2, out, n);
}